// SVAE_LDS_86088324481265
// MI455X (gfx1250) — compile-verified
//
#include <hip/hip_runtime.h>
#include <math.h>

#define BB   32
#define TT   256
#define OBS  32
#define LAT  64
#define HID  400
#define NROW (BB * TT)

typedef float v2f __attribute__((ext_vector_type(2)));
typedef float v8f __attribute__((ext_vector_type(8)));

// ---------------------------------------------------------------------------
// WMMA f32 16x16x4 wrapper + fragment loaders (CDNA5 wave32 layouts)
// A 16x4: lanes 0-15 rows M=0..15 {K,K+1}, lanes 16-31 same rows {K+2,K+3}
// B 4x16: lanes 0-15 cols N=0..15 rows {K,K+1}, lanes 16-31 rows {K+2,K+3}
// C/D:    VGPR r = row r (lanes 0-15) / row r+8 (lanes 16-31), col = lane%16
// ---------------------------------------------------------------------------
__device__ __forceinline__ v8f wmma4(v2f a, v2f b, v8f c) {
  return __builtin_amdgcn_wmma_f32_16x16x4_f32(false, a, false, b, (short)0, c,
                                               false, false);
}

__device__ __forceinline__ v2f afrag(const float* A, int ld, int r0, int k0,
                                     int lane) {
  int r = r0 + (lane & 15);
  int k = k0 + ((lane >> 4) << 1);
  v2f a;
  a.x = A[r * ld + k];
  a.y = A[r * ld + k + 1];
  return a;
}
__device__ __forceinline__ v2f bfrag(const float* B, int ld, int k0, int c0,
                                     int lane) {
  int c = c0 + (lane & 15);
  int k = k0 + ((lane >> 4) << 1);
  v2f b;
  b.x = B[k * ld + c];
  b.y = B[(k + 1) * ld + c];
  return b;
}
// B^T fragment: (B^T)[k][c] = B[c][k]
__device__ __forceinline__ v2f bfragT(const float* B, int ld, int k0, int c0,
                                      int lane) {
  int c = c0 + (lane & 15);
  int k = k0 + ((lane >> 4) << 1);
  v2f b;
  b.x = B[c * ld + k];
  b.y = B[c * ld + k + 1];
  return b;
}
// (I - K) fragments, fused from K in LDS (Joseph form)
__device__ __forceinline__ v2f afrag_ImK(const float* K, int r0, int k0,
                                         int lane) {
  int r = r0 + (lane & 15);
  int k = k0 + ((lane >> 4) << 1);
  v2f a;
  a.x = ((r == k) ? 1.f : 0.f) - K[r * LAT + k];
  a.y = ((r == k + 1) ? 1.f : 0.f) - K[r * LAT + k + 1];
  return a;
}
__device__ __forceinline__ v2f bfrag_ImKT(const float* K, int k0, int c0,
                                          int lane) {
  int c = c0 + (lane & 15);
  int k = k0 + ((lane >> 4) << 1);
  v2f b;
  b.x = ((c == k) ? 1.f : 0.f) - K[c * LAT + k];
  b.y = ((c == k + 1) ? 1.f : 0.f) - K[c * LAT + k + 1];
  return b;
}
// K * diag(Rd) fragment (columns of K scaled by Rd)
__device__ __forceinline__ v2f afrag_scaled(const float* K, const float* s,
                                            int r0, int k0, int lane) {
  int r = r0 + (lane & 15);
  int k = k0 + ((lane >> 4) << 1);
  v2f a;
  a.x = K[r * LAT + k] * s[k];
  a.y = K[r * LAT + k + 1] * s[k + 1];
  return a;
}

// ---------------------------------------------------------------------------
// Async global -> LDS copy of a 64x64 f32 matrix (16 KB), 256 threads.
// Each lane issues four B128 async transfers; tracked by ASYNCcnt.
// ---------------------------------------------------------------------------
__device__ __forceinline__ void async_copy_mat(float* lds_dst,
                                               const float* gsrc, int tid) {
#pragma unroll
  for (int i = 0; i < 4; ++i) {
    int f = (tid + i * 256) * 4;  // float index
    unsigned laddr = (unsigned)(uintptr_t)lds_dst + (unsigned)(f * 4);
    const float* g = gsrc + f;
    asm volatile("global_load_async_to_lds_b128 %0, %1, off" ::"v"(laddr),
                 "v"(g)
                 : "memory");
  }
}
__device__ __forceinline__ void wait_async0() {
  asm volatile("s_wait_asynccnt 0x0" ::: "memory");
}

// Synchronous vectorized copy helpers (B128 path)
__device__ __forceinline__ void copy_mat_v4(float* dst, const float* src,
                                            int tid, int nthr) {
  const float4* s = (const float4*)src;
  float4* d = (float4*)dst;
  for (int i = tid; i < 1024; i += nthr) d[i] = s[i];
}

// ---------------------------------------------------------------------------
// 64x64 matmuls in LDS: 16 tiles, 8 waves -> 2 tiles/wave, K=16 wmma steps
// ---------------------------------------------------------------------------
__device__ __forceinline__ void mm64(const float* A, const float* B,
                                     float* Cout, const float* addM, int wave,
                                     int lane) {
  for (int t = wave; t < 16; t += 8) {
    int r0 = (t >> 2) << 4, c0 = (t & 3) << 4;
    v8f acc = {0, 0, 0, 0, 0, 0, 0, 0};
#pragma unroll
    for (int kk = 0; kk < 16; ++kk)
      acc = wmma4(afrag(A, LAT, r0, kk * 4, lane),
                  bfrag(B, LAT, kk * 4, c0, lane), acc);
    int c = c0 + (lane & 15);
    int rb = r0 + ((lane >> 4) << 3);
#pragma unroll
    for (int r = 0; r < 8; ++r) {
      float v = acc[r];
      if (addM) v += addM[(rb + r) * LAT + c];
      Cout[(rb + r) * LAT + c] = v;
    }
  }
}
__device__ __forceinline__ void mm64_bt(const float* A, const float* B,
                                        float* Cout, const float* addM,
                                        int wave, int lane) {
  for (int t = wave; t < 16; t += 8) {
    int r0 = (t >> 2) << 4, c0 = (t & 3) << 4;
    v8f acc = {0, 0, 0, 0, 0, 0, 0, 0};
#pragma unroll
    for (int kk = 0; kk < 16; ++kk)
      acc = wmma4(afrag(A, LAT, r0, kk * 4, lane),
                  bfragT(B, LAT, kk * 4, c0, lane), acc);
    int c = c0 + (lane & 15);
    int rb = r0 + ((lane >> 4) << 3);
#pragma unroll
    for (int r = 0; r < 8; ++r) {
      float v = acc[r];
      if (addM) v += addM[(rb + r) * LAT + c];
      Cout[(rb + r) * LAT + c] = v;
    }
  }
}

// Cooperative in-place Cholesky of a 64x64 SPD matrix in LDS.
__device__ __forceinline__ void chol64(float* M, int tid, int nthr) {
  for (int j = 0; j < 64; ++j) {
    __syncthreads();
    if (tid == 0) M[j * 65] = sqrtf(M[j * 65]);
    __syncthreads();
    float d = M[j * 65];
    if (tid > j && tid < 64) M[tid * 64 + j] /= d;
    __syncthreads();
    for (int e = tid; e < 4096; e += nthr) {
      int rr = e >> 6, cc = e & 63;
      if (cc > j && cc <= rr) M[e] -= M[rr * 64 + j] * M[cc * 64 + j];
    }
  }
  __syncthreads();
}

// ---------------------------------------------------------------------------
// Kernel 1: Q = M M^T from packed lower-tri params (exp on diag)
// ---------------------------------------------------------------------------
__global__ void __launch_bounds__(256) buildQ_kernel(const float* q, float* Q) {
  __shared__ float M[4096];
  int tid = threadIdx.x;
  for (int e = tid; e < 4096; e += 256) {
    int i = e >> 6, j = e & 63;
    float v = 0.f;
    if (j < i)
      v = q[i * (i + 1) / 2 + j];
    else if (j == i)
      v = expf(q[i * (i + 1) / 2 + i]);
    M[e] = v;
  }
  __syncthreads();
  for (int e = tid; e < 4096; e += 256) {
    int i = e >> 6, j = e & 63;
    int kmax = (i < j) ? i : j;
    float s = 0.f;
    for (int k = 0; k <= kmax; ++k) s += M[i * 64 + k] * M[j * 64 + k];
    Q[e] = s;
  }
}

// ---------------------------------------------------------------------------
// Kernel 2: h = relu(x @ W1 + b1)   (8192x32 @ 32x400)
// ---------------------------------------------------------------------------
__global__ void __launch_bounds__(256) enc1_kernel(const float* x,
                                                   const float* W1,
                                                   const float* b1, float* h) {
  int tid = threadIdx.x, wave = tid >> 5, lane = tid & 31;
  int m0 = blockIdx.x * 16;
  for (int nt = wave; nt < 25; nt += 8) {
    int n0 = nt * 16;
    v8f acc = {0, 0, 0, 0, 0, 0, 0, 0};
#pragma unroll
    for (int kk = 0; kk < 8; ++kk)
      acc = wmma4(afrag(x, OBS, m0, kk * 4, lane),
                  bfrag(W1, HID, kk * 4, n0, lane), acc);
    int c = n0 + (lane & 15);
    int rb = m0 + ((lane >> 4) << 3);
    float bias = b1[c];
#pragma unroll
    for (int r = 0; r < 8; ++r) {
      float v = acc[r] + bias;
      h[(size_t)(rb + r) * HID + c] = (v > 0.f) ? v : 0.f;
    }
  }
}

// ---------------------------------------------------------------------------
// Kernel 3: z_mu = h@Wm + bm ; sigma = exp(0.5*(h@Wl + bl))
// ---------------------------------------------------------------------------
__global__ void __launch_bounds__(256) enc2_kernel(const float* h,
                                                   const float* Wm,
                                                   const float* bm,
                                                   const float* Wl,
                                                   const float* bl, float* zmu,
                                                   float* sg) {
  int tid = threadIdx.x, wave = tid >> 5, lane = tid & 31;
  int m0 = blockIdx.x * 16;
  const float* B = (wave < 4) ? Wm : Wl;
  int n0 = (wave & 3) * 16;
  v8f acc = {0, 0, 0, 0, 0, 0, 0, 0};
#pragma unroll 4
  for (int kk = 0; kk < 100; ++kk)
    acc = wmma4(afrag(h, HID, m0, kk * 4, lane),
                bfrag(B, LAT, kk * 4, n0, lane), acc);
  int c = n0 + (lane & 15);
  int rb = m0 + ((lane >> 4) << 3);
  if (wave < 4) {
    float bias = bm[c];
#pragma unroll
    for (int r = 0; r < 8; ++r)
      zmu[(size_t)(rb + r) * LAT + c] = acc[r] + bias;
  } else {
    float bias = bl[c];
#pragma unroll
    for (int r = 0; r < 8; ++r)
      sg[(size_t)(rb + r) * LAT + c] = expf(0.5f * (acc[r] + bias));
  }
}

// ---------------------------------------------------------------------------
// Kernel 4: Kalman filter. One block per sequence; state in LDS; sequential T.
// Writes mf into mq region, Sf into Sq region (smoother fixes them in place).
// ---------------------------------------------------------------------------
__global__ void __launch_bounds__(256) filter_kernel(
    const float* zmu, const float* Rg, const int* mask, const float* Ag,
    const float* bvec, const float* Qg, float* out_mf, float* out_Sf,
    float* out_ll) {
  extern __shared__ float sm[];
  float* sA = sm;           // 4096  A
  float* sSig = sA + 4096;  // 4096  Sig (state)
  float* s1 = sSig + 4096;  // 4096  AS -> K
  float* s2 = s1 + 4096;    // 4096  Sig_p
  float* s3 = s2 + 4096;    // 4096  S/C -> T1
  float* vmu = s3 + 4096;   // 64
  float* vmp = vmu + 64;
  float* vr = vmp + 64;
  float* vRd = vr + 64;
  float* vmu_u = vRd + 64;
  float* vb = vmu_u + 64;

  int tid = threadIdx.x, wave = tid >> 5, lane = tid & 31;
  int b = blockIdx.x;
  copy_mat_v4(sA, Ag, tid, 256);
  for (int e = tid; e < 4096; e += 256)
    sSig[e] = ((e >> 6) == (e & 63)) ? 1.f : 0.f;
  if (tid < 64) {
    vmu[tid] = 0.f;
    vb[tid] = bvec[tid];
  }
  float llacc = 0.f;  // thread 0 only
  __syncthreads();

  for (int t = 0; t < TT; ++t) {
    size_t row = (size_t)b * TT + t;
    // R diag, mu_p = A@mu + b, innovation r = y - mu_p
    if (tid < 64) {
      vRd[tid] = Rg[row * LAT + tid];
      float s = vb[tid];
      for (int k = 0; k < 64; ++k) s += sA[tid * 64 + k] * vmu[k];
      vmp[tid] = s;
      vr[tid] = zmu[row * LAT + tid] - s;
    }
    __syncthreads();
    mm64(sA, sSig, s1, nullptr, wave, lane);  // s1 = A @ Sig
    __syncthreads();
    mm64_bt(s1, sA, s2, Qg, wave, lane);  // s2 = Sig_p = A Sig A^T + Q
    __syncthreads();
    // s3 = S = Sig_p + diag(R); then Cholesky in place -> C
    for (int e = tid; e < 4096; e += 256) {
      int rr = e >> 6, cc = e & 63;
      s3[e] = s2[e] + ((rr == cc) ? vRd[rr] : 0.f);
    }
    chol64(s3, tid, 256);
    // K = (S^{-1} Sig_p)^T into s1: thread c solves RHS column c, writes row c
    if (tid < 64) {
      int c = tid;
      for (int i = 0; i < 64; ++i) {
        float v = s2[i * 64 + c];
        for (int k = 0; k < i; ++k) v -= s3[i * 64 + k] * s1[c * 64 + k];
        s1[c * 64 + i] = v / s3[i * 65];
      }
      for (int i = 63; i >= 0; --i) {
        float v = s1[c * 64 + i];
        for (int k = i + 1; k < 64; ++k) v -= s3[k * 64 + i] * s1[c * 64 + k];
        s1[c * 64 + i] = v / s3[i * 65];
      }
    }
    __syncthreads();
    // mu_u = mu_p + K r
    if (tid < 64) {
      float s = 0.f;
      for (int k = 0; k < 64; ++k) s += s1[tid * 64 + k] * vr[k];
      vmu_u[tid] = vmp[tid] + s;
    }
    int masked = mask[row];
    if (tid == 0 && masked == 0) {  // log-likelihood via C alpha = r
      float al[64];
      float sll = -32.f * 1.8378770664093453f;  // -0.5*LAT*log(2pi)
      for (int i = 0; i < 64; ++i) {
        float v = vr[i];
        for (int k = 0; k < i; ++k) v -= s3[i * 64 + k] * al[k];
        float dii = s3[i * 65];
        al[i] = v / dii;
        sll -= logf(dii);
        sll -= 0.5f * al[i] * al[i];
      }
      llacc += sll;
    }
    __syncthreads();
    // T1 = (I-K) @ Sig_p -> s3 (C is dead)
    for (int tt2 = wave; tt2 < 16; tt2 += 8) {
      int r0 = (tt2 >> 2) << 4, c0 = (tt2 & 3) << 4;
      v8f acc = {0, 0, 0, 0, 0, 0, 0, 0};
#pragma unroll
      for (int kk = 0; kk < 16; ++kk)
        acc = wmma4(afrag_ImK(s1, r0, kk * 4, lane),
                    bfrag(s2, LAT, kk * 4, c0, lane), acc);
      int c = c0 + (lane & 15);
      int rb = r0 + ((lane >> 4) << 3);
#pragma unroll
      for (int r = 0; r < 8; ++r) s3[(rb + r) * LAT + c] = acc[r];
    }
    __syncthreads();
    // Sig_u = T1 (I-K)^T + (K R) K^T ; select vs Sig_p; store state + output
    for (int tt2 = wave; tt2 < 16; tt2 += 8) {
      int r0 = (tt2 >> 2) << 4, c0 = (tt2 & 3) << 4;
      v8f acc = {0, 0, 0, 0, 0, 0, 0, 0};
#pragma unroll
      for (int kk = 0; kk < 16; ++kk)
        acc = wmma4(afrag(s3, LAT, r0, kk * 4, lane),
                    bfrag_ImKT(s1, kk * 4, c0, lane), acc);
#pragma unroll
      for (int kk = 0; kk < 16; ++kk)
        acc = wmma4(afrag_scaled(s1, vRd, r0, kk * 4, lane),
                    bfragT(s1, LAT, kk * 4, c0, lane), acc);
      int c = c0 + (lane & 15);
      int rb = r0 + ((lane >> 4) << 3);
#pragma unroll
      for (int r = 0; r < 8; ++r) {
        float v = masked ? s2[(rb + r) * LAT + c] : acc[r];
        sSig[(rb + r) * LAT + c] = v;
        out_Sf[row * 4096 + (size_t)(rb + r) * LAT + c] = v;
      }
    }
    if (tid < 64) {
      float v = masked ? vmp[tid] : vmu_u[tid];
      vmu[tid] = v;
      out_mf[row * LAT + tid] = v;
    }
    __syncthreads();
  }
  if (tid == 0) out_ll[b] = llacc;
}

// ---------------------------------------------------------------------------
// Kernel 5: RTS smoother, backward in place over mq/Sq.
// Double-buffered sF with ASYNC global->LDS prefetch of the next step's tile.
// ---------------------------------------------------------------------------
__global__ void __launch_bounds__(256) smoother_kernel(const float* Ag,
                                                       const float* bvec,
                                                       const float* Qg,
                                                       float* mq, float* Sq) {
  extern __shared__ float sm[];
  float* sA = sm;              // A
  float* sS = sA + 4096;       // Sig_s carry
  float* sF0 = sS + 4096;      // Sig_f buffer 0
  float* sF1 = sF0 + 4096;     // Sig_f buffer 1
  float* s1 = sF1 + 4096;      // A@Sf
  float* s2 = s1 + 4096;       // Sig_pred -> D
  float* s3 = s2 + 4096;       // C -> G@D
  float* s4 = s3 + 4096;       // G
  float* vmu_s = s4 + 4096;    // 64
  float* vmu_f = vmu_s + 64;   // 64
  float* vmp = vmu_f + 64;     // 64
  float* vtmp = vmp + 64;      // 64
  float* vb = vtmp + 64;       // 64

  int tid = threadIdx.x, wave = tid >> 5, lane = tid & 31;
  int b = blockIdx.x;
  copy_mat_v4(sA, Ag, tid, 256);
  if (tid < 64) vb[tid] = bvec[tid];
  size_t rowT = (size_t)b * TT + (TT - 1);
  copy_mat_v4(sS, Sq + rowT * 4096, tid, 256);
  if (tid < 64) vmu_s[tid] = mq[rowT * LAT + tid];
  // Prefetch Sf[T-2] into buffer 0 (ASYNCcnt-tracked).
  async_copy_mat(sF0, Sq + ((size_t)b * TT + (TT - 2)) * 4096, tid);
  __syncthreads();

  for (int t = TT - 2; t >= 0; --t) {
    size_t row = (size_t)b * TT + t;
    int i = (TT - 2 - t);
    float* sF = (i & 1) ? sF1 : sF0;
    float* sFn = (i & 1) ? sF0 : sF1;
    wait_async0();  // prefetched Sf(t) is now resident
    __syncthreads();
    if (t > 0) async_copy_mat(sFn, Sq + (row - 1) * 4096, tid);
    if (tid < 64) vmu_f[tid] = mq[row * LAT + tid];
    __syncthreads();
    if (tid < 64) {  // mu_pred = A mu_f + b
      float s = vb[tid];
      for (int k = 0; k < 64; ++k) s += sA[tid * 64 + k] * vmu_f[k];
      vmp[tid] = s;
    }
    mm64(sA, sF, s1, nullptr, wave, lane);  // s1 = A @ Sf
    __syncthreads();
    mm64_bt(s1, sA, s2, Qg, wave, lane);  // s2 = Sig_pred
    __syncthreads();
    for (int e = tid; e < 4096; e += 256) s3[e] = s2[e];
    chol64(s3, tid, 256);
    // G = (Sig_pred^{-1} A Sf)^T into s4
    if (tid < 64) {
      int c = tid;
      for (int i2 = 0; i2 < 64; ++i2) {
        float v = s1[i2 * 64 + c];
        for (int k = 0; k < i2; ++k) v -= s3[i2 * 64 + k] * s4[c * 64 + k];
        s4[c * 64 + i2] = v / s3[i2 * 65];
      }
      for (int i2 = 63; i2 >= 0; --i2) {
        float v = s4[c * 64 + i2];
        for (int k = i2 + 1; k < 64; ++k) v -= s3[k * 64 + i2] * s4[c * 64 + k];
        s4[c * 64 + i2] = v / s3[i2 * 65];
      }
    }
    __syncthreads();
    // D = Sig_s - Sig_pred (into s2)
    for (int e = tid; e < 4096; e += 256) s2[e] = sS[e] - s2[e];
    __syncthreads();
    mm64(s4, s2, s3, nullptr, wave, lane);  // s3 = G @ D
    __syncthreads();
    // Sig = Sf + (G D) G^T -> carry + output
    for (int tt2 = wave; tt2 < 16; tt2 += 8) {
      int r0 = (tt2 >> 2) << 4, c0 = (tt2 & 3) << 4;
      v8f acc = {0, 0, 0, 0, 0, 0, 0, 0};
#pragma unroll
      for (int kk = 0; kk < 16; ++kk)
        acc = wmma4(afrag(s3, LAT, r0, kk * 4, lane),
                    bfragT(s4, LAT, kk * 4, c0, lane), acc);
      int c = c0 + (lane & 15);
      int rb = r0 + ((lane >> 4) << 3);
#pragma unroll
      for (int r = 0; r < 8; ++r) {
        float v = acc[r] + sF[(rb + r) * LAT + c];
        sS[(rb + r) * LAT + c] = v;
        Sq[row * 4096 + (size_t)(rb + r) * LAT + c] = v;
      }
    }
    // mu = mu_f + G (mu_s - mu_pred)
    if (tid < 64) {
      float s = 0.f;
      for (int k = 0; k < 64; ++k) s += s4[tid * 64 + k] * (vmu_s[k] - vmp[k]);
      vtmp[tid] = vmu_f[tid] + s;
    }
    __syncthreads();
    if (tid < 64) {
      vmu_s[tid] = vtmp[tid];
      mq[row * LAT + tid] = vtmp[tid];
    }
    __syncthreads();
  }
}

// ---------------------------------------------------------------------------
// Kernel 6: per-(b,t) Cholesky of Sq + 1e-5 I, z = mq + L eps, x = z Wd + bd
// ---------------------------------------------------------------------------
__global__ void __launch_bounds__(64) sample_decode_kernel(
    const float* eps, const float* Wd, const float* bd, const float* mq,
    const float* Sq, float* xr) {
  __shared__ float L[4096];
  __shared__ float z[64];
  int tid = threadIdx.x;
  size_t row = blockIdx.x;
  copy_mat_v4(L, Sq + row * 4096, tid, 64);
  __syncthreads();
  if (tid < 64) L[tid * 65] += 1e-5f;
  chol64(L, tid, 64);
  if (tid < 64) {
    float s = mq[row * LAT + tid];
    for (int k = 0; k <= tid; ++k) s += L[tid * 64 + k] * eps[row * LAT + k];
    z[tid] = s;
  }
  __syncthreads();
  if (tid < 32) {
    float s = bd[tid];
    for (int k = 0; k < 64; ++k) s += z[k] * Wd[k * OBS + tid];
    xr[row * OBS + tid] = s;
  }
}

// ---------------------------------------------------------------------------
extern "C" void kernel_launch(void* const* d_in, const int* in_sizes, int n_in,
                              void* d_out, int out_size, void* d_ws,
                              size_t ws_size, hipStream_t stream) {
  const float* x = (const float*)d_in[0];
  const int* mask = (const int*)d_in[1];
  const float* eps = (const float*)d_in[2];
  const float* W1 = (const float*)d_in[3];
  const float* b1 = (const float*)d_in[4];
  const float* Wm = (const float*)d_in[5];
  const float* bm = (const float*)d_in[6];
  const float* Wl = (const float*)d_in[7];
  const float* bl = (const float*)d_in[8];
  const float* Wd = (const float*)d_in[9];
  const float* bd = (const float*)d_in[10];
  const float* A = (const float*)d_in[11];
  const float* bvec = (const float*)d_in[12];
  const float* qpar = (const float*)d_in[13];

  float* out = (float*)d_out;
  float* xr = out;           // 32*256*32 = 262144
  float* ll = out + 262144;  // 32
  float* mq = out + 262176;  // 32*256*64 = 524288
  float* Sq = out + 786464;  // 32*256*64*64 = 33554432

  float* ws = (float*)d_ws;
  float* h = ws;  // 8192*400
  float* zmu = h + (size_t)NROW * HID;
  float* sg = zmu + (size_t)NROW * LAT;
  float* Q = sg + (size_t)NROW * LAT;  // 4096

  buildQ_kernel<<<1, 256, 0, stream>>>(qpar, Q);
  enc1_kernel<<<NROW / 16, 256, 0, stream>>>(x, W1, b1, h);
  enc2_kernel<<<NROW / 16, 256, 0, stream>>>(h, Wm, bm, Wl, bl, zmu, sg);

  size_t smemF = (5 * 4096 + 6 * 64) * sizeof(float);  // ~82 KB
  size_t smemS = (8 * 4096 + 5 * 64) * sizeof(float);  // ~129 KB
  (void)hipFuncSetAttribute((const void*)filter_kernel,
                            hipFuncAttributeMaxDynamicSharedMemorySize,
                            (int)smemF);
  (void)hipFuncSetAttribute((const void*)smoother_kernel,
                            hipFuncAttributeMaxDynamicSharedMemorySize,
                            (int)smemS);

  filter_kernel<<<BB, 256, smemF, stream>>>(zmu, sg, mask, A, bvec, Q, mq, Sq,
                                            ll);
  smoother_kernel<<<BB, 256, smemS, stream>>>(A, bvec, Q, mq, Sq);
  sample_decode_kernel<<<NROW, 64, 0, stream>>>(eps, Wd, bd, mq, Sq, xr);
}